// DWT_188978561034
// MI455X (gfx1250) — compile-verified
//
#include <hip/hip_runtime.h>
#include <cstdint>

// 2x2 Haar DWT, (16,64,256,256) f32 -> (16,256,128,128) f32.
// Memory-bound (0.5 FLOP/byte): design goal is pure 23.3 TB/s streaming.
// Uses gfx1250 async global->LDS DMA (ASYNCcnt) to double-buffer 16KB tiles.

#define THREADS          256
#define TILES_PER_BLOCK  8
#define PLANES           1024          // B*C = 16*64
#define TILES_PER_PLANE  16            // 256 rows / 16 rows-per-tile
#define TILE_FLOATS      4096          // 16 rows * 256 cols
#define OUT_BAND_STRIDE  16384         // 128*128

typedef __attribute__((ext_vector_type(4))) float v4f;   // true clang vector: OK for NT builtins

__device__ __forceinline__ void async_ld_b128(uint32_t lds_byte_off, const float* gptr) {
    // vdst = VGPR holding per-lane LDS byte address, vaddr = 64-bit global addr
    asm volatile("global_load_async_to_lds_b128 %0, %1, off"
                 :: "v"(lds_byte_off), "v"(gptr)
                 : "memory");
}

#define HAAR(aa, bb, cc, dd, A, Hh, Vv, Dd)            \
    {                                                  \
        float s0 = (aa) + (bb), s1 = (cc) + (dd);      \
        float t0 = (bb) - (aa), t1 = (dd) - (cc);      \
        (A)  = (s0 + s1) * 0.5f;                       \
        (Hh) = (s1 - s0) * 0.5f;                       \
        (Vv) = (t0 + t1) * 0.5f;                       \
        (Dd) = (t1 - t0) * 0.5f;                       \
    }

__global__ __launch_bounds__(THREADS)
void DWT_haar_kernel(const float* __restrict__ x, float* __restrict__ out) {
    __shared__ __align__(16) float smem[2][TILE_FLOATS];   // 32 KB double buffer

    const uint32_t tid = threadIdx.x;
    // Low 32 bits of a flat shared-aperture address == LDS offset (ISA 10.2).
    const uint32_t lds_base = (uint32_t)(uintptr_t)(&smem[0][0]);

    const int g0 = (int)blockIdx.x * TILES_PER_BLOCK;

    // Issue 4 x b128 async copies per thread: one contiguous 16KB tile.
    auto prefetch = [&](int g, int bufsel) {
        const float* src = x + ((size_t)g * (size_t)TILE_FLOATS);
        const uint32_t lb = lds_base + (uint32_t)bufsel * (TILE_FLOATS * 4u);
#pragma unroll
        for (uint32_t j = 0; j < 4; ++j) {
            const uint32_t e = j * 256u + tid;     // float4 index within tile
            async_ld_b128(lb + e * 16u, src + (size_t)e * 4u);
        }
    };

    prefetch(g0, 0);

    const int wave = (int)(tid >> 5);   // output row within tile: 0..7
    const int lane = (int)(tid & 31);   // group of 4 output cols: 0..31

    for (int i = 0; i < TILES_PER_BLOCK; ++i) {
        if (i + 1 < TILES_PER_BLOCK) {
            prefetch(g0 + i + 1, (i + 1) & 1);
            // 4 newer async ops may be in flight; oldest 4 (tile i) must be done.
            asm volatile("s_wait_asynccnt 4" ::: "memory");
        } else {
            asm volatile("s_wait_asynccnt 0" ::: "memory");
        }
        __syncthreads();   // tile i visible to all waves

        const int g     = g0 + i;
        const int plane = g >> 4;   // / TILES_PER_PLANE
        const int tp    = g & 15;   // % TILES_PER_PLANE

        const float* bse = &smem[i & 1][0];
        const int r0 = wave * 512 + lane * 8;      // 2*wave rows of 256 floats
        const v4f u0 = *(const v4f*)(bse + r0);
        const v4f u1 = *(const v4f*)(bse + r0 + 4);
        const v4f v0 = *(const v4f*)(bse + r0 + 256);
        const v4f v1 = *(const v4f*)(bse + r0 + 260);

        v4f cA, cH, cV, cD;
        HAAR(u0.x, u0.y, v0.x, v0.y, cA.x, cH.x, cV.x, cD.x);
        HAAR(u0.z, u0.w, v0.z, v0.w, cA.y, cH.y, cV.y, cD.y);
        HAAR(u1.x, u1.y, v1.x, v1.y, cA.z, cH.z, cV.z, cD.z);
        HAAR(u1.z, u1.w, v1.z, v1.w, cA.w, cH.w, cV.w, cD.w);

        const size_t orow = (size_t)(tp * 8 + wave);
        const size_t ob   = ((size_t)plane * 4u) * (size_t)OUT_BAND_STRIDE
                          + orow * 128u + (size_t)lane * 4u;

        // One float4 per sub-band; coalesced, streaming (non-temporal) stores.
        __builtin_nontemporal_store(cA, (v4f*)(out + ob));
        __builtin_nontemporal_store(cH, (v4f*)(out + ob + OUT_BAND_STRIDE));
        __builtin_nontemporal_store(cV, (v4f*)(out + ob + 2 * OUT_BAND_STRIDE));
        __builtin_nontemporal_store(cD, (v4f*)(out + ob + 3 * OUT_BAND_STRIDE));

        __syncthreads();   // reads done before this buffer is overwritten
    }
}

extern "C" void kernel_launch(void* const* d_in, const int* in_sizes, int n_in,
                              void* d_out, int out_size, void* d_ws, size_t ws_size,
                              hipStream_t stream) {
    (void)in_sizes; (void)n_in; (void)d_ws; (void)ws_size; (void)out_size;
    const float* x = (const float*)d_in[0];
    float* out = (float*)d_out;

    const int total_tiles = PLANES * TILES_PER_PLANE;       // 16384
    const int blocks = total_tiles / TILES_PER_BLOCK;       // 2048
    DWT_haar_kernel<<<blocks, THREADS, 0, stream>>>(x, out);
}